// PersonalizationLayer_30528627540712
// MI455X (gfx1250) — compile-verified
//
#include <hip/hip_runtime.h>
#include <cstdint>
#include <cstddef>

#ifndef __has_builtin
#define __has_builtin(x) 0
#endif

// Probe-confirmed (round 1): both builtins AVAILABLE on this toolchain.
#if __has_builtin(__builtin_amdgcn_global_load_async_to_lds_b128)
#define HAVE_ASYNC_LDS 1
#else
#define HAVE_ASYNC_LDS 0
#endif

#if __has_builtin(__builtin_amdgcn_s_wait_asynccnt)
#define WAIT_ASYNCCNT0() __builtin_amdgcn_s_wait_asynccnt(0)
#else
#define WAIT_ASYNCCNT0() asm volatile("s_wait_asynccnt 0" ::: "memory")
#endif

namespace {

constexpr int   kH     = 16;                      // head dim (reference: H=16)
constexpr int   kBlock = 256;                     // 8 wave32
constexpr float kEps   = 1e-7f;
constexpr float kLog2e = 1.4426950408889634f;     // log2(e)

// Builtin signature (from round-1 diagnostic): pointers to 16B int vectors.
typedef int v4i __attribute__((ext_vector_type(4)));
typedef __attribute__((address_space(1))) v4i* gptr_v4i;
typedef __attribute__((address_space(3))) v4i* lptr_v4i;

// sigmoid(scale*logit(p) + bias), all in base-2 hardware trans ops:
//   logit(p) = ln(p/(1-p)) = ln2 * (log2(p) - log2(1-p))
//   sigmoid(z) = 1 / (1 + exp2(-z*log2e))
//   -z*log2e = -(s*(log2 p - log2(1-p)) + bias*log2e)   [c = bias*log2e]
__device__ __forceinline__ float pers1(float x, float s, float c) {
  float p  = __builtin_fminf(__builtin_fmaxf(x, kEps), 1.0f - kEps);
  float l2 = __builtin_amdgcn_logf(p) - __builtin_amdgcn_logf(1.0f - p); // v_log_f32 x2
  float q  = __builtin_fmaf(-s, l2, -c);                                 // v_fma_f32
  float e  = __builtin_amdgcn_exp2f(q);                                  // v_exp_f32
  return __builtin_amdgcn_rcpf(1.0f + e);                                // v_rcp_f32
}

} // namespace

__global__ __launch_bounds__(kBlock) void personalization_kernel(
    const float* __restrict__ probs,
    const float* __restrict__ scale_table,
    const float* __restrict__ bias_table,
    const int*   __restrict__ user_idx,
    float* __restrict__ out,
    int sh4) // (S*H)/4 float4 elements per batch row (== 800)
{
  const int b = blockIdx.x;
  const int t = threadIdx.x;
  const int u = user_idx[b]; // uniform per block -> scalar load

  __shared__ __align__(16) float s_scale[kH];
  __shared__ __align__(16) float s_bias[kH];

#if HAVE_ASYNC_LDS
  // Stage the 64B scale row + 64B bias row with async B128 copies (gfx1250 path):
  // lanes 0..3 -> scale row, lanes 4..7 -> bias row, 16 bytes per lane.
  if (t < 8) {
    const float* g = (t < 4) ? (scale_table + (size_t)u * kH + 4 * t)
                             : (bias_table  + (size_t)u * kH + 4 * (t - 4));
    float* l = (t < 4) ? &s_scale[4 * t] : &s_bias[4 * (t - 4)];
    __builtin_amdgcn_global_load_async_to_lds_b128(
        (gptr_v4i)g, (lptr_v4i)l, 0 /*offset*/, 0 /*cpol*/);
  }
  WAIT_ASYNCCNT0();
  __syncthreads();
#else
  if (t < kH) {
    s_scale[t] = scale_table[(size_t)u * kH + t];
    s_bias[t]  = bias_table[(size_t)u * kH + t];
  }
  __syncthreads();
#endif

  // Each float4 chunk i covers h = (4*i) mod 16 .. +3; with stride 256 (1024
  // elements, multiple of 16) this h-slice is loop-invariant per thread.
  const int h0 = (t << 2) & (kH - 1);
  const float4 s4 = *reinterpret_cast<const float4*>(&s_scale[h0]);
  const float4 b4 = *reinterpret_cast<const float4*>(&s_bias[h0]);
  const float4 c4 = make_float4(b4.x * kLog2e, b4.y * kLog2e,
                                b4.z * kLog2e, b4.w * kLog2e);

  const float4* __restrict__ pin  =
      reinterpret_cast<const float4*>(probs) + (size_t)b * sh4;
  float4* __restrict__ pout =
      reinterpret_cast<float4*>(out) + (size_t)b * sh4;

  for (int i = t; i < sh4; i += kBlock) {
    __builtin_prefetch(pin + i + kBlock, 0, 0); // global_prefetch_b8 (speculative)
    const float4 x = pin[i];                    // global_load_b128
    float4 r;
    r.x = pers1(x.x, s4.x, c4.x);
    r.y = pers1(x.y, s4.y, c4.y);
    r.z = pers1(x.z, s4.z, c4.z);
    r.w = pers1(x.w, s4.w, c4.w);
    pout[i] = r;                                // global_store_b128
  }
}

extern "C" void kernel_launch(void* const* d_in, const int* in_sizes, int n_in,
                              void* d_out, int out_size, void* d_ws, size_t ws_size,
                              hipStream_t stream) {
  (void)n_in; (void)out_size; (void)d_ws; (void)ws_size;
  const float* probs       = (const float*)d_in[0];
  const float* scale_table = (const float*)d_in[1];
  const float* bias_table  = (const float*)d_in[2];
  const int*   user_idx    = (const int*)d_in[3];
  float*       out         = (float*)d_out;

  const int B   = in_sizes[3];          // 4096
  const int sh  = in_sizes[0] / B;      // S*H = 3200
  const int sh4 = sh / 4;               // 800 float4 chunks per row

  personalization_kernel<<<dim3(B), dim3(kBlock), 0, stream>>>(
      probs, scale_table, bias_table, user_idx, out, sh4);
}